// MHParentPredictor_66005057405235
// MI455X (gfx1250) — compile-verified
//
#include <hip/hip_runtime.h>
#include <hip/hip_bf16.h>

typedef float v2f __attribute__((ext_vector_type(2)));
typedef float v8f __attribute__((ext_vector_type(8)));

#define N_EDGES 262144
#define C_CLS   2048
#define PD      128
#define MD      128
#define HDIM    128
#define P_OUT   512

// ---- workspace layout (float offsets) ----
#define WS_WTP    0                        // packed W_pdt^T : 128x128
#define WS_WTC    (WS_WTP + 128*128)       // packed W_comb^T: 256x128
#define WS_WTO    (WS_WTC + 256*128)       // packed W_out^T : 128x512
#define WS_SCORES (WS_WTO + 128*512)       // N scores
#define WS_PMAX   (WS_SCORES + N_EDGES)    // 1024 block maxima
#define WS_PSUM   (WS_PMAX + 1024)         // 1024 block sums
#define WS_MS     (WS_PSUM + 1024)         // global {M, S}

__device__ __forceinline__ v8f splat8(float x) {
    v8f r;
#pragma unroll
    for (int i = 0; i < 8; ++i) r[i] = x;
    return r;
}

// ---------------------------------------------------------------------------
// 8 independent 16x16 accumulator tiles over a K=128 panel.
// arow: &tile[lr0+am][ak]   (A fragment base for this lane, LDS)
// wt:   packed weight base already offset by (ah*stride2 + 2*am + 32*ntbase)
// stride2 = 2*Ncols (floats per packed k-pair row)
// Per ks: 1 ds_load_b64 + 8 coalesced global_load_b64 + 8 independent v_wmma.
// ---------------------------------------------------------------------------
__device__ __forceinline__ void gemm_block8(const float* __restrict__ arow,
                                            const float* __restrict__ wt,
                                            int stride2, v8f acc[8]) {
    for (int ks = 0; ks < 32; ++ks) {
        v2f a = *(const v2f*)(arow + 4 * ks);
        const float* base = wt + (size_t)(2 * ks) * stride2;
#pragma unroll
        for (int t = 0; t < 8; ++t) {
            v2f b = *(const v2f*)(base + t * 32);
            acc[t] = __builtin_amdgcn_wmma_f32_16x16x4_f32(false, a, false, b,
                                                           (short)0, acc[t], false, false);
        }
    }
}

// ---------------------------------------------------------------------------
// Kernel 1: transpose + pack weights: B[k][n] (= W[n][k]) stored at
// (k>>1)*(2*Ncols) + 2*n + (k&1)  ->  one lane's two K-values = one b64 load.
// ---------------------------------------------------------------------------
__global__ __launch_bounds__(256) void prep_weights(const float* __restrict__ Wp,
                                                    const float* __restrict__ Wc,
                                                    const float* __restrict__ Wo,
                                                    float* __restrict__ ws) {
    int i = blockIdx.x * 256 + threadIdx.x;   // 0 .. 114687
    if (i < 128*128) {                        // W_pdt [H,PD]: K=PD, Ncols=H
        int k = i >> 7, n = i & 127;
        ws[WS_WTP + (k >> 1) * 256 + 2 * n + (k & 1)] = Wp[n * 128 + k];
    } else if (i < 128*128 + 256*128) {       // W_comb [H,256]: K=256, Ncols=H
        int j = i - 128*128;
        int k = j >> 7, n = j & 127;
        ws[WS_WTC + (k >> 1) * 256 + 2 * n + (k & 1)] = Wc[n * 256 + k];
    } else {                                  // W_out [P,H]: K=H, Ncols=P
        int j = i - (128*128 + 256*128);
        int k = j >> 9, n = j & 511;
        ws[WS_WTO + (k >> 1) * 1024 + 2 * n + (k & 1)] = Wo[n * 128 + k];
    }
}

// ---------------------------------------------------------------------------
// Kernel 2: per-edge attention score + per-block softmax partials (max, sumexp)
// ---------------------------------------------------------------------------
__global__ __launch_bounds__(256) void score_kernel(const float* __restrict__ pf,
                                                    const float* __restrict__ cls,
                                                    const float* __restrict__ Wa,
                                                    const float* __restrict__ ba,
                                                    const int*   __restrict__ ei,
                                                    float* __restrict__ ws) {
    __shared__ float red[256];
    int tid = threadIdx.x;
    size_t e = (size_t)blockIdx.x * 256 + tid;

    float s = ba[0];
    const float* pr = pf + e * PD;
#pragma unroll 8
    for (int c = 0; c < PD; c += 4) {
        float4 p = *(const float4*)(pr + c);
        float4 w = *(const float4*)(Wa + c);
        s += p.x * w.x + p.y * w.y + p.z * w.z + p.w * w.w;
    }
    int cidx = ei[N_EDGES + e];               // edge_index[1][e]
    const float* cr = cls + (size_t)cidx * MD;
#pragma unroll 8
    for (int c = 0; c < MD; c += 4) {
        float4 p = *(const float4*)(cr + c);
        float4 w = *(const float4*)(Wa + PD + c);
        s += p.x * w.x + p.y * w.y + p.z * w.z + p.w * w.w;
    }
    ws[WS_SCORES + e] = s;

    red[tid] = s;
    __syncthreads();
    for (int off = 128; off > 0; off >>= 1) {
        if (tid < off) red[tid] = fmaxf(red[tid], red[tid + off]);
        __syncthreads();
    }
    float bmax = red[0];
    __syncthreads();
    red[tid] = __expf(s - bmax);
    __syncthreads();
    for (int off = 128; off > 0; off >>= 1) {
        if (tid < off) red[tid] += red[tid + off];
        __syncthreads();
    }
    if (tid == 0) {
        ws[WS_PMAX + blockIdx.x] = bmax;
        ws[WS_PSUM + blockIdx.x] = red[0];
    }
}

// ---------------------------------------------------------------------------
// Kernel 3: combine 1024 partials -> global (M, S)
// ---------------------------------------------------------------------------
__global__ __launch_bounds__(1024) void combine_kernel(float* __restrict__ ws) {
    __shared__ float rmax[1024];
    __shared__ float rsum[1024];
    int tid = threadIdx.x;
    float m = ws[WS_PMAX + tid];
    rmax[tid] = m;
    __syncthreads();
    for (int off = 512; off > 0; off >>= 1) {
        if (tid < off) rmax[tid] = fmaxf(rmax[tid], rmax[tid + off]);
        __syncthreads();
    }
    float M = rmax[0];
    __syncthreads();
    rsum[tid] = ws[WS_PSUM + tid] * __expf(m - M);
    __syncthreads();
    for (int off = 512; off > 0; off >>= 1) {
        if (tid < off) rsum[tid] += rsum[tid + off];
        __syncthreads();
    }
    if (tid == 0) { ws[WS_MS] = M; ws[WS_MS + 1] = rsum[0]; }
}

// ---------------------------------------------------------------------------
// Kernel 4: fused 3-stage GEMM chain, fp32 WMMA, 8-way interleaved acc tiles.
// 64 rows/block, 4 waves, 16 rows/wave. LDS stride 132 -> conflict-free frags.
// ---------------------------------------------------------------------------
__global__ __launch_bounds__(128) void fused_kernel(const float* __restrict__ pf,
                                                    const float* __restrict__ cls,
                                                    const int*   __restrict__ ei,
                                                    const float* __restrict__ bp,
                                                    const float* __restrict__ bc,
                                                    const float* __restrict__ bo,
                                                    const float* __restrict__ ws,
                                                    float* __restrict__ out) {
    __shared__ float tileA[64][132];   // pf tile, later pdt tile (per-wave rows)
    __shared__ float tileB[64][132];   // w*cf tile, later comb tile (per-wave rows)
    __shared__ float wrow[64];

    const float* WtP = ws + WS_WTP;
    const float* WtC = ws + WS_WTC;
    const float* WtO = ws + WS_WTO;
    const float* scores = ws + WS_SCORES;

    int tid  = threadIdx.x;
    int lane = tid & 31;
    int wv   = tid >> 5;
    size_t row0 = (size_t)blockIdx.x * 64;

    float M = ws[WS_MS], S = ws[WS_MS + 1];
    if (tid < 64) wrow[tid] = __expf(scores[row0 + tid] - M) / S;

    for (int i = tid * 4; i < 64 * 128; i += 128 * 4) {
        int r = i >> 7, c = i & 127;
        float4 v = *(const float4*)(pf + (row0 + r) * (size_t)PD + c);
        tileA[r][c] = v.x; tileA[r][c + 1] = v.y; tileA[r][c + 2] = v.z; tileA[r][c + 3] = v.w;
    }
    __syncthreads();   // wrow ready
    for (int i = tid * 4; i < 64 * 128; i += 128 * 4) {
        int r = i >> 7, c = i & 127;
        int cidx = ei[N_EDGES + row0 + r];
        float w = wrow[r];
        float4 v = *(const float4*)(cls + (size_t)cidx * MD + c);
        tileB[r][c] = w * v.x; tileB[r][c + 1] = w * v.y;
        tileB[r][c + 2] = w * v.z; tileB[r][c + 3] = w * v.w;
    }
    __syncthreads();

    const int lr0 = wv * 16;
    const int am  = lane & 15;     // M (A) / N (B,D) index within tile
    const int ah  = lane >> 4;     // half: K offset (A,B) / M offset 0|8 (D)
    const int ak  = ah * 2;

    const float* arowA = &tileA[lr0 + am][ak];
    const float* arowB = &tileB[lr0 + am][ak];

    v8f acc[8];

    // ---- Stage 1: pdt = pf @ W_pdt^T + b_pdt  (K=128, Nout=128) ----
#pragma unroll
    for (int t = 0; t < 8; ++t) acc[t] = splat8(bp[t * 16 + am]);
    gemm_block8(arowA, WtP + ah * 256 + 2 * am, 256, acc);
    // pdt overwrites this wave's pf rows (own-wave rows; LDS in-order in-wave)
#pragma unroll
    for (int t = 0; t < 8; ++t)
#pragma unroll
        for (int v = 0; v < 8; ++v)
            tileA[lr0 + ah * 8 + v][t * 16 + am] = acc[t][v];

    // ---- Stage 2: comb = pdt@Wc1^T + agg@Wc2^T + b_comb  (K=256, Nout=128) ----
#pragma unroll
    for (int t = 0; t < 8; ++t) acc[t] = splat8(bc[t * 16 + am]);
    gemm_block8(arowA, WtC + ah * 256 + 2 * am, 256, acc);             // K 0..127 (pdt)
    gemm_block8(arowB, WtC + 64 * 256 + ah * 256 + 2 * am, 256, acc);  // K 128..255 (agg)
    // comb overwrites this wave's agg rows
#pragma unroll
    for (int t = 0; t < 8; ++t)
#pragma unroll
        for (int v = 0; v < 8; ++v)
            tileB[lr0 + ah * 8 + v][t * 16 + am] = acc[t][v];

    // ---- Stage 3: logits = comb @ W_out^T + b_out  (K=128, Nout=512) ----
    for (int g = 0; g < 4; ++g) {
#pragma unroll
        for (int t = 0; t < 8; ++t) acc[t] = splat8(bo[(g * 8 + t) * 16 + am]);
        gemm_block8(arowB, WtO + ah * 1024 + 2 * am + 32 * (g * 8), 1024, acc);
        size_t grow = row0 + lr0 + ah * 8;
#pragma unroll
        for (int t = 0; t < 8; ++t) {
            int col = (g * 8 + t) * 16 + am;
#pragma unroll
            for (int v = 0; v < 8; ++v)
                out[(grow + v) * (size_t)P_OUT + col] = acc[t][v];
        }
    }
}

// ---------------------------------------------------------------------------
extern "C" void kernel_launch(void* const* d_in, const int* in_sizes, int n_in,
                              void* d_out, int out_size, void* d_ws, size_t ws_size,
                              hipStream_t stream) {
    const float* pf  = (const float*)d_in[0];
    const float* cls = (const float*)d_in[1];
    const float* Wa  = (const float*)d_in[2];
    const float* ba  = (const float*)d_in[3];
    const float* Wp  = (const float*)d_in[4];
    const float* bp  = (const float*)d_in[5];
    const float* Wc  = (const float*)d_in[6];
    const float* bc  = (const float*)d_in[7];
    const float* Wo  = (const float*)d_in[8];
    const float* bo  = (const float*)d_in[9];
    const int*   ei  = (const int*)d_in[10];
    float* out = (float*)d_out;
    float* ws  = (float*)d_ws;

    prep_weights<<<(128*128 + 256*128 + 128*512) / 256, 256, 0, stream>>>(Wp, Wc, Wo, ws);
    score_kernel<<<N_EDGES / 256, 256, 0, stream>>>(pf, cls, Wa, ba, ei, ws);
    combine_kernel<<<1, 1024, 0, stream>>>(ws);
    fused_kernel<<<N_EDGES / 64, 128, 0, stream>>>(pf, cls, ei, bp, bc, bo, ws, out);
}